// GNN_13589276524753
// MI455X (gfx1250) — compile-verified
//
#include <hip/hip_runtime.h>

typedef __attribute__((ext_vector_type(2))) float v2f;
typedef __attribute__((ext_vector_type(8))) float v8f;

#define KNN 10
#define LRELU(x) ((x) > 0.0f ? (x) : 0.2f * (x))

// ---------------------------------------------------------------------------
// Kernel 1: transpose [B,C,N] -> [B,N,C] and squared norms per point.
// ---------------------------------------------------------------------------
__global__ void prep_kernel(const float* __restrict__ x,   // [B, C, N]
                            float* __restrict__ xt,        // [B, N, C]
                            float* __restrict__ sq,        // [B, N]
                            int N, int C) {
    int n = blockIdx.x * blockDim.x + threadIdx.x;
    int b = blockIdx.y;
    if (n >= N) return;
    const float* src = x + ((size_t)b * C) * N + n;
    float* dst = xt + ((size_t)b * N + n) * C;
    float s = 0.0f;
    for (int c = 0; c < C; ++c) {
        float v = src[(size_t)c * N];
        s += v * v;
        dst[c] = v;
    }
    sq[(size_t)b * N + n] = s;
}

// ---------------------------------------------------------------------------
// Kernel 2: brute-force kNN (k=10), C = 64, one wave32 per 16-row tile.
//  - A tile (16 rows x K=64) resident in registers, reused for all columns.
//  - Two 16-col tiles per iteration: B fragments pre-loaded into register
//    arrays (loads clause together), two independent WMMA accumulator chains.
//  - Next iteration's B loads issued BEFORE the top-k scan (dead-register
//    software pipeline) so the scan hides global-load latency.
//  - Scan uses all 32 lanes (each row scanned by two lanes, one per column
//    half); per-row sorted top-10 lists merged once at the end with
//    value/index tie-break (matches jax.lax.top_k ordering).
// ---------------------------------------------------------------------------
__global__ __launch_bounds__(32) void knn_kernel(
    const float* __restrict__ xt,   // [B, N, 64]
    const float* __restrict__ sq,   // [B, N]
    int* __restrict__ idx,          // [B, N, KNN]
    int N) {
    const int C = 64;
    const int ntiles = N >> 4;      // 16-col tiles
    const int nct = ntiles >> 1;    // 32-col iterations
    int b    = blockIdx.x / ntiles;
    int row0 = (blockIdx.x % ntiles) << 4;
    int lane = threadIdx.x;
    int l15  = lane & 15;
    int koff = (lane < 16) ? 0 : 2;
    int half = lane >> 4;           // which 16-col half this lane scans

    __shared__ float tileLds[512];  // two 16x16 inner-product tiles
    __shared__ float sqcol[32];
    __shared__ float mv[32 * KNN];  // per-lane top-10 values for final merge
    __shared__ int   mi[32 * KNN];  // per-lane top-10 indices

    // A tile: rows row0..row0+15, full K=64, register resident
    const float* arow = xt + ((size_t)b * N + row0 + l15) * C;
    v2f A[16];
#pragma unroll
    for (int kk = 0; kk < 16; ++kk) {
        A[kk].x = arow[4 * kk + koff];
        A[kk].y = arow[4 * kk + koff + 1];
    }
    // Each lane scans row row0 + l15 (two lanes per row, one per col-half)
    float sqr = sq[(size_t)b * N + row0 + l15];

    float tv[KNN];
    int   ti[KNN];
#pragma unroll
    for (int q = 0; q < KNN; ++q) { tv[q] = -3.0e38f; ti[q] = 0; }

    v2f B0[16], B1[16];
    auto loadB = [&](int ct) {
        int col0 = ct << 5;
        const float* br0 = xt + ((size_t)b * N + col0 + l15) * C;
        const float* br1 = br0 + 16 * C;
#pragma unroll
        for (int kk = 0; kk < 16; ++kk) {
            B0[kk].x = br0[4 * kk + koff];
            B0[kk].y = br0[4 * kk + koff + 1];
            B1[kk].x = br1[4 * kk + koff];
            B1[kk].y = br1[4 * kk + koff + 1];
        }
    };

    loadB(0);
    for (int ct = 0; ct < nct; ++ct) {
        int col0 = ct << 5;
        // Two independent accumulator chains over the shared A tile
        v8f acc0 = {};
        v8f acc1 = {};
#pragma unroll
        for (int kk = 0; kk < 16; ++kk) {
            acc0 = __builtin_amdgcn_wmma_f32_16x16x4_f32(
                false, A[kk], false, B0[kk], (short)0, acc0, false, false);
            acc1 = __builtin_amdgcn_wmma_f32_16x16x4_f32(
                false, A[kk], false, B1[kk], (short)0, acc1, false, false);
        }
        // Spill both tiles to LDS for the per-row scan
        int trow = half ? 8 : 0;
#pragma unroll
        for (int r = 0; r < 8; ++r) {
            tileLds[(trow + r) * 16 + l15]       = acc0[r];
            tileLds[256 + (trow + r) * 16 + l15] = acc1[r];
        }
        sqcol[lane] = sq[(size_t)b * N + col0 + lane];
        // B0/B1 are dead now: kick off next iteration's loads before the
        // scan so the scan hides their latency.
        if (ct + 1 < nct) loadB(ct + 1);
        __syncthreads();
        {
            const float* rowp = tileLds + half * 256 + l15 * 16;
            int cbase = col0 + half * 16;
#pragma unroll 4
            for (int j = 0; j < 16; ++j) {
                float v = 2.0f * rowp[j] - sqr - sqcol[half * 16 + j];
                if (v > tv[KNN - 1]) {
                    tv[KNN - 1] = v;
                    ti[KNN - 1] = cbase + j;
#pragma unroll
                    for (int p = KNN - 1; p > 0; --p) {
                        if (tv[p] > tv[p - 1]) {
                            float fv = tv[p]; tv[p] = tv[p - 1]; tv[p - 1] = fv;
                            int   iv = ti[p]; ti[p] = ti[p - 1]; ti[p - 1] = iv;
                        }
                    }
                }
            }
        }
        __syncthreads();
    }

    // Merge the two column-half lists per row (stable: value, then index)
#pragma unroll
    for (int q = 0; q < KNN; ++q) {
        mv[lane * KNN + q] = tv[q];
        mi[lane * KNN + q] = ti[q];
    }
    __syncthreads();
    if (lane < 16) {
        int p0 = lane * KNN;
        int p1 = (lane + 16) * KNN;
        int* op = idx + ((size_t)b * N + row0 + lane) * KNN;
#pragma unroll
        for (int q = 0; q < KNN; ++q) {
            float v0 = mv[p0], v1 = mv[p1];
            int   i0 = mi[p0], i1 = mi[p1];
            bool take0 = (v0 > v1) || (v0 == v1 && i0 < i1);
            op[q] = take0 ? i0 : i1;
            if (take0) ++p0; else ++p1;
        }
    }
}

// ---------------------------------------------------------------------------
// Kernel 3: EdgeConv. One block per point, one thread per output channel.
// edge = concat(x_j - x_i, x_i); W:[Cout, 2*Cin].
// Center term (bias + (W_hi - W_lo)·x_i) hoisted out of the k-loop.
// ---------------------------------------------------------------------------
__global__ void edgeconv_kernel(const float* __restrict__ xt_in,  // [B,N,Cin]
                                const int* __restrict__ idx,      // [B,N,KNN]
                                const float* __restrict__ W,      // [Cout, 2*Cin]
                                const float* __restrict__ bias,   // [Cout]
                                float* __restrict__ xt_out,       // [B,N,Cout]
                                float* __restrict__ sq_out,       // [B,N] or null
                                int N, int Cin, int Cout) {
    __shared__ float sh[64 + 64 + 128];   // xi | xj | reduction
    float* xi  = sh;
    float* xj  = sh + 64;
    float* red = sh + 128;

    int bn = blockIdx.x;          // b*N + n
    int o  = threadIdx.x;         // output channel
    int b  = bn / N;

    const float* center = xt_in + (size_t)bn * Cin;
    for (int c = o; c < Cin; c += blockDim.x) xi[c] = center[c];
    __syncthreads();

    const float* wrow = W + (size_t)o * 2 * Cin;
    float cacc = bias[o];
#pragma unroll 8
    for (int c = 0; c < Cin; ++c) cacc += (wrow[Cin + c] - wrow[c]) * xi[c];

    float hmax = -3.0e38f;
    const int* nb = idx + (size_t)bn * KNN;
    for (int k = 0; k < KNN; ++k) {
        __syncthreads();
        int j = nb[k];
        const float* nbr = xt_in + ((size_t)b * N + j) * Cin;
        for (int c = o; c < Cin; c += blockDim.x) xj[c] = nbr[c];
        __syncthreads();
        float acc = cacc;
#pragma unroll 8
        for (int c = 0; c < Cin; ++c) acc += wrow[c] * xj[c];
        acc = LRELU(acc);
        hmax = fmaxf(hmax, acc);
    }
    xt_out[(size_t)bn * Cout + o] = hmax;

    if (sq_out != nullptr) {
        red[o] = hmax * hmax;
        __syncthreads();
        for (int s = blockDim.x >> 1; s > 0; s >>= 1) {
            if (o < s) red[o] += red[o + s];
            __syncthreads();
        }
        if (o == 0) sq_out[bn] = red[0];
    }
}

// ---------------------------------------------------------------------------
// Kernel 4: final pointwise conv over virtual concat(x0,x1,x2) (K=256),
// out[b,o,n] = leaky(W3[o,:]·cat + b3[o]). WMMA GEMM, 16x16 output tiles.
// K-loop batched in groups of 8 so A/B fragment loads clause together.
// ---------------------------------------------------------------------------
__global__ __launch_bounds__(32) void pointwise_kernel(
    const float* __restrict__ xt0,   // [B,N,64]
    const float* __restrict__ xt1,   // [B,N,64]
    const float* __restrict__ xt2,   // [B,N,128]
    const float* __restrict__ W3,    // [64, 256]
    const float* __restrict__ b3,    // [64]
    float* __restrict__ out,         // [B, 64, N]
    int N) {
    const int ntiles = N >> 4;
    int blk = blockIdx.x;
    int nt  = blk % ntiles;
    int ot  = (blk / ntiles) & 3;
    int b   = blk / (ntiles * 4);
    int o0  = ot << 4;
    int n0  = nt << 4;
    int lane = threadIdx.x;
    int l15  = lane & 15;
    int koff = (lane < 16) ? 0 : 2;

    const float* arow = W3 + (size_t)(o0 + l15) * 256;
    v8f acc = {};
    // 8 groups of 8 K-steps; each group's 32 channels lie in one source
    // buffer (boundaries at c=64,128 are multiples of 32).
#pragma unroll
    for (int g = 0; g < 8; ++g) {
        int cg = g * 32;                 // first channel of this group
        const float* src;
        int cc, cw;
        if (cg < 64)       { src = xt0; cc = cg;       cw = 64;  }
        else if (cg < 128) { src = xt1; cc = cg - 64;  cw = 64;  }
        else               { src = xt2; cc = cg - 128; cw = 128; }
        const float* bp = src + ((size_t)b * N + n0 + l15) * cw + cc + koff;
        const float* ap = arow + cg + koff;
        v2f Av[8], Bv[8];
#pragma unroll
        for (int u = 0; u < 8; ++u) {
            Av[u].x = ap[4 * u];
            Av[u].y = ap[4 * u + 1];
            Bv[u].x = bp[4 * u];
            Bv[u].y = bp[4 * u + 1];
        }
#pragma unroll
        for (int u = 0; u < 8; ++u) {
            acc = __builtin_amdgcn_wmma_f32_16x16x4_f32(
                false, Av[u], false, Bv[u], (short)0, acc, false, false);
        }
    }
    int mh = (lane < 16) ? 0 : 8;
#pragma unroll
    for (int r = 0; r < 8; ++r) {
        int o = o0 + r + mh;
        float v = acc[r] + b3[o];
        v = LRELU(v);
        out[((size_t)b * 64 + o) * N + n0 + l15] = v;
    }
}

// ---------------------------------------------------------------------------
// Host-side launch
// ---------------------------------------------------------------------------
extern "C" void kernel_launch(void* const* d_in, const int* in_sizes, int n_in,
                              void* d_out, int out_size, void* d_ws, size_t ws_size,
                              hipStream_t stream) {
    (void)in_sizes; (void)n_in; (void)out_size; (void)ws_size;
    const float* x0 = (const float*)d_in[0];
    const float* W1 = (const float*)d_in[1];
    const float* b1 = (const float*)d_in[2];
    const float* W2 = (const float*)d_in[3];
    const float* b2 = (const float*)d_in[4];
    const float* W3 = (const float*)d_in[5];
    const float* b3 = (const float*)d_in[6];
    const int B = 2, F = 64, N = 8192;

    // Workspace partition (256B-aligned regions), total ~17.3 MB
    char* ws = (char*)d_ws;
    size_t off = 0;
    auto takef = [&](size_t elems) {
        void* p = ws + off;
        off += (elems * 4 + 255) & ~(size_t)255;
        return p;
    };
    float* xt0 = (float*)takef((size_t)B * N * F);
    float* xt1 = (float*)takef((size_t)B * N * F);
    float* xt2 = (float*)takef((size_t)B * N * 2 * F);
    float* sq0 = (float*)takef((size_t)B * N);
    float* sq1 = (float*)takef((size_t)B * N);
    int*  idx1 = (int*)takef((size_t)B * N * KNN);
    int*  idx2 = (int*)takef((size_t)B * N * KNN);

    dim3 pgrid((N + 255) / 256, B);
    prep_kernel<<<pgrid, 256, 0, stream>>>(x0, xt0, sq0, N, F);

    knn_kernel<<<B * (N / 16), 32, 0, stream>>>(xt0, sq0, idx1, N);
    edgeconv_kernel<<<B * N, F, 0, stream>>>(xt0, idx1, W1, b1, xt1, sq1, N, F, F);

    knn_kernel<<<B * (N / 16), 32, 0, stream>>>(xt1, sq1, idx2, N);
    edgeconv_kernel<<<B * N, 2 * F, 0, stream>>>(xt1, idx2, W2, b2, xt2, nullptr, N, F, 2 * F);

    pointwise_kernel<<<B * 4 * (N / 16), 32, 0, stream>>>(
        xt0, xt1, xt2, W3, b3, (float*)d_out, N);
}